// SynAlign_24043226923583
// MI455X (gfx1250) — compile-verified
//
#include <hip/hip_runtime.h>
#include <math.h>

#define NEG_INF -9999.0f

constexpr int B = 64, S = 128, T = 128, D = 1024;

typedef __attribute__((ext_vector_type(2))) float v2f;
typedef __attribute__((ext_vector_type(8))) float v8f;

// gfx1250 async global->LDS copy path (ASYNCcnt), with sync fallback.
#if __has_builtin(__builtin_amdgcn_global_load_async_to_lds_b128) && \
    __has_builtin(__builtin_amdgcn_s_wait_asynccnt)
#define USE_ASYNC_COPY 1
#define AS1 __attribute__((address_space(1)))
#define AS3 __attribute__((address_space(3)))
typedef int vi4 __attribute__((vector_size(16)));
#else
#define USE_ASYNC_COPY 0
#endif

__device__ __forceinline__ void copy16(const float* __restrict__ g,
                                       float* __restrict__ l) {
#if USE_ASYNC_COPY
  __builtin_amdgcn_global_load_async_to_lds_b128(
      (AS1 vi4*)(void*)g, (AS3 vi4*)(void*)l, 0, 0);
#else
  *(float4*)l = *(const float4*)g;
#endif
}

__device__ __forceinline__ void copy_fence() {
#if USE_ASYNC_COPY
  __builtin_amdgcn_s_wait_asynccnt(0);
#endif
}

// ---------------------------------------------------------------------------
// Kernel A: embedding gather + avg-pool(3, SAME, edge-count=2) + residual.
// One block per (b, l) row; 256 threads x float4 covers D=1024.
// ---------------------------------------------------------------------------
__global__ void embed_pool_kernel(const float* __restrict__ table,
                                  const int* __restrict__ sent,
                                  float* __restrict__ out, int L) {
  int row = blockIdx.x;            // b*L + l
  int l = row % L;
  int idx_c = sent[row];
  int idx_m = (l > 0)     ? sent[row - 1] : -1;
  int idx_p = (l < L - 1) ? sent[row + 1] : -1;
  float cnt = 1.0f + (idx_m >= 0 ? 1.0f : 0.0f) + (idx_p >= 0 ? 1.0f : 0.0f);
  float inv = 1.0f / cnt;

  const float4* ec = (const float4*)(table + (size_t)idx_c * D);
  const float4* em = (idx_m >= 0) ? (const float4*)(table + (size_t)idx_m * D) : nullptr;
  const float4* ep = (idx_p >= 0) ? (const float4*)(table + (size_t)idx_p * D) : nullptr;
  float4* o = (float4*)(out + (size_t)row * D);

  int d = threadIdx.x;             // 0..255 -> float4 index
  float4 c = ec[d];
  float sx = c.x, sy = c.y, sz = c.z, sw = c.w;
  if (em) { float4 m = em[d]; sx += m.x; sy += m.y; sz += m.z; sw += m.w; }
  if (ep) { float4 p = ep[d]; sx += p.x; sy += p.y; sz += p.z; sw += p.w; }
  float4 r;
  r.x = sx * inv + c.x;
  r.y = sy * inv + c.y;
  r.z = sz * inv + c.z;
  r.w = sw * inv + c.w;
  o[d] = r;
}

// ---------------------------------------------------------------------------
// Kernel B: mul_score[b] = tgt_sm[b] (T x D) @ src_sm[b]^T (D x S) -> [T,S]
// fp32 WMMA 16x16x4 with LDS-staged 32-wide K panels (stride 34 = conflict-
// free, 8B-aligned). One block per batch; 8 waves; wave w owns T-tile-row w.
// ---------------------------------------------------------------------------
__global__ void score_kernel(const float* __restrict__ tgt_sm,
                             const float* __restrict__ src_sm,
                             float* __restrict__ score) {
  constexpr int KB = 32, LDSS = 34;
  __shared__ float lT[128 * LDSS];   // tgt panel  [t][kk]
  __shared__ float lS[128 * LDSS];   // src panel  [s][kk]

  int b    = blockIdx.x;
  int tid  = threadIdx.x;
  int wave = tid >> 5;
  int lane = tid & 31;
  int half = lane >> 4;            // 0 or 1
  int l16  = lane & 15;
  int trow = wave * 16 + l16;      // A-matrix row (M)
  int koff = half * 2;             // A/B K sub-offset per ISA layout

  const float* Tg = tgt_sm + (size_t)b * T * D;
  const float* Sg = src_sm + (size_t)b * S * D;

  int r0 = tid >> 3;               // 0..31 (staging row within pass)
  int c4 = (tid & 7) * 4;          // 0..28 (staging float4 column)

  v8f acc[8] = {};
  for (int kp = 0; kp < D; kp += KB) {
    __syncthreads();               // previous panel fully consumed
#pragma unroll
    for (int p = 0; p < 4; ++p) {  // 4 passes x 32 rows = 128 rows
      int r = p * 32 + r0;
      copy16(Tg + (size_t)r * D + kp + c4, &lT[r * LDSS + c4]);
      copy16(Sg + (size_t)r * D + kp + c4, &lS[r * LDSS + c4]);
    }
    copy_fence();
    __syncthreads();

    for (int kk = 0; kk < KB; kk += 4) {
      v2f a = *(const v2f*)&lT[trow * LDSS + kk + koff];
      v2f bb[8];
#pragma unroll
      for (int tn = 0; tn < 8; ++tn) {
        int s = tn * 16 + l16;
        bb[tn] = *(const v2f*)&lS[s * LDSS + kk + koff];
      }
#pragma unroll
      for (int tn = 0; tn < 8; ++tn)
        acc[tn] = __builtin_amdgcn_wmma_f32_16x16x4_f32(
            false, a, false, bb[tn], (short)0, acc[tn], false, false);
    }
  }

  // C/D layout: VGPR r -> row (r + half*8), col = l16.
  int rbase = wave * 16 + half * 8;
  float* outb = score + (size_t)b * T * S;
#pragma unroll
  for (int tn = 0; tn < 8; ++tn) {
    int col = tn * 16 + l16;
#pragma unroll
    for (int r = 0; r < 8; ++r)
      outb[(size_t)(rbase + r) * S + col] = acc[tn][r];
  }
}

// ---------------------------------------------------------------------------
// Kernel C1: ta_soft = softmax_S( mask ? score[b,t,s] : NEG_INF ) -> [B,T,S]
// ---------------------------------------------------------------------------
__global__ void softmax_ts_kernel(const float* __restrict__ score,
                                  const int* __restrict__ src_sent,
                                  const int* __restrict__ tgt_sent,
                                  float* __restrict__ out) {
  int b = blockIdx.x / T;
  int t = blockIdx.x % T;
  int s = threadIdx.x;             // 128 threads
  bool valid = (src_sent[b * S + s] > 0) && (tgt_sent[b * T + t] > 0);
  float v = score[((size_t)b * T + t) * S + s];
  float x = valid ? v : NEG_INF;

  __shared__ float red[128];
  red[s] = x;
  __syncthreads();
  for (int off = 64; off > 0; off >>= 1) {
    if (s < off) red[s] = fmaxf(red[s], red[s + off]);
    __syncthreads();
  }
  float mx = red[0];
  __syncthreads();
  float e = expf(x - mx);
  red[s] = e;
  __syncthreads();
  for (int off = 64; off > 0; off >>= 1) {
    if (s < off) red[s] += red[s + off];
    __syncthreads();
  }
  out[((size_t)b * T + t) * S + s] = e / red[0];
}

// ---------------------------------------------------------------------------
// Kernel C2: at_soft = softmax_T( mask ? score[b,t,s] : NEG_INF ) -> [B,S,T]
// ---------------------------------------------------------------------------
__global__ void softmax_st_kernel(const float* __restrict__ score,
                                  const int* __restrict__ src_sent,
                                  const int* __restrict__ tgt_sent,
                                  float* __restrict__ out) {
  int b = blockIdx.x / S;
  int s = blockIdx.x % S;
  int t = threadIdx.x;             // 128 threads
  bool valid = (src_sent[b * S + s] > 0) && (tgt_sent[b * T + t] > 0);
  float v = score[((size_t)b * T + t) * S + s];
  float x = valid ? v : NEG_INF;

  __shared__ float red[128];
  red[t] = x;
  __syncthreads();
  for (int off = 64; off > 0; off >>= 1) {
    if (t < off) red[t] = fmaxf(red[t], red[t + off]);
    __syncthreads();
  }
  float mx = red[0];
  __syncthreads();
  float e = expf(x - mx);
  red[t] = e;
  __syncthreads();
  for (int off = 64; off > 0; off >>= 1) {
    if (t < off) red[t] += red[t + off];
    __syncthreads();
  }
  out[((size_t)b * S + s) * T + t] = e / red[0];
}

// ---------------------------------------------------------------------------
// Kernel D: O[b] = P[b] (128 x 128) @ V[b] (128 x D), fp32 WMMA 16x16x4.
// grid = (B, D/128). LDS-staged: P panel [128 x 34], V panel [32 x 128]
// (lane-consecutive N reads -> conflict-free).
// ---------------------------------------------------------------------------
__global__ void attn_value_kernel(const float* __restrict__ P,
                                  const float* __restrict__ V,
                                  float* __restrict__ O) {
  constexpr int M = 128, K = 128, KB = 32, LDSS = 34;
  __shared__ float lP[128 * LDSS];   // P panel [m][kk]
  __shared__ float lV[KB * 128];     // V panel [kk][n]

  int b    = blockIdx.x;
  int nc   = blockIdx.y * 128;
  int tid  = threadIdx.x;
  int wave = tid >> 5;
  int lane = tid & 31;
  int half = lane >> 4;
  int l16  = lane & 15;
  int mrow = wave * 16 + l16;
  int koff = half * 2;

  const float* Pb = P + (size_t)b * M * K;
  const float* Vb = V + (size_t)b * K * D;
  float* Ob = O + (size_t)b * M * D;

  int r0  = tid >> 3;              // P staging: 32 rows/pass
  int c4  = (tid & 7) * 4;
  int rv0 = tid >> 5;              // V staging: 8 rows/pass
  int cv4 = (tid & 31) * 4;

  v8f acc[8] = {};
  for (int kp = 0; kp < K; kp += KB) {
    __syncthreads();
#pragma unroll
    for (int p = 0; p < 4; ++p) {
      int rp = p * 32 + r0;
      copy16(Pb + (size_t)rp * K + kp + c4, &lP[rp * LDSS + c4]);
      int rv = p * 8 + rv0;
      copy16(Vb + (size_t)(kp + rv) * D + nc + cv4, &lV[rv * 128 + cv4]);
    }
    copy_fence();
    __syncthreads();

    for (int kk = 0; kk < KB; kk += 4) {
      v2f a = *(const v2f*)&lP[mrow * LDSS + kk + koff];
      v2f bb[8];
#pragma unroll
      for (int nt = 0; nt < 8; ++nt) {
        int n = nt * 16 + l16;
        v2f t;
        t.x = lV[(kk + koff) * 128 + n];
        t.y = lV[(kk + koff + 1) * 128 + n];
        bb[nt] = t;
      }
#pragma unroll
      for (int nt = 0; nt < 8; ++nt)
        acc[nt] = __builtin_amdgcn_wmma_f32_16x16x4_f32(
            false, a, false, bb[nt], (short)0, acc[nt], false, false);
    }
  }

  int rbase = wave * 16 + half * 8;
#pragma unroll
  for (int nt = 0; nt < 8; ++nt) {
    int n = nc + nt * 16 + l16;
#pragma unroll
    for (int r = 0; r < 8; ++r)
      Ob[(size_t)(rbase + r) * D + n] = acc[nt][r];
  }
}

// ---------------------------------------------------------------------------
// Launch. Output tuple (flat, return order):
//   out0 src            [B,S,D]  @ 0
//   out1 source_att     [B,T,D]  @ 8388608
//   out2 at_soft_score  [B,S,T]  @ 16777216
//   out3 tgt            [B,T,D]  @ 17825792
//   out4 target_att     [B,S,D]  @ 26214400
//   out5 ta_soft_score  [B,T,S]  @ 34603008
// ---------------------------------------------------------------------------
extern "C" void kernel_launch(void* const* d_in, const int* in_sizes, int n_in,
                              void* d_out, int out_size, void* d_ws, size_t ws_size,
                              hipStream_t stream) {
  const int* src_sent = (const int*)d_in[0];
  const int* tgt_sent = (const int*)d_in[1];
  // d_in[2]/d_in[3] (bool masks) intentionally unused: masks are recomputed
  // from sent > 0, matching the reference and avoiding bool-dtype ambiguity.
  const float* src_tab = (const float*)d_in[4];
  const float* tgt_tab = (const float*)d_in[5];

  float* out = (float*)d_out;
  float* out_src  = out;
  float* out_satt = out + 8388608;
  float* out_at   = out + 16777216;
  float* out_tgt  = out + 17825792;
  float* out_tatt = out + 26214400;
  float* out_ta   = out + 34603008;

  float* ws_score = (float*)d_ws;   // B*T*S fp32 = 4 MiB raw scores

  embed_pool_kernel<<<B * S, 256, 0, stream>>>(src_tab, src_sent, out_src, S);
  embed_pool_kernel<<<B * T, 256, 0, stream>>>(tgt_tab, tgt_sent, out_tgt, T);

  score_kernel<<<B, 256, 0, stream>>>(out_tgt, out_src, ws_score);

  softmax_ts_kernel<<<B * T, 128, 0, stream>>>(ws_score, src_sent, tgt_sent, out_ta);
  softmax_st_kernel<<<B * S, 128, 0, stream>>>(ws_score, src_sent, tgt_sent, out_at);

  attn_value_kernel<<<dim3(B, 8), 256, 0, stream>>>(out_ta, out_src, out_satt);
  attn_value_kernel<<<dim3(B, 8), 256, 0, stream>>>(out_at, out_tgt, out_tatt);
}